// Linear_47055661695233
// MI455X (gfx1250) — compile-verified
//
#include <hip/hip_runtime.h>

typedef float v2f __attribute__((ext_vector_type(2)));
typedef float v8f __attribute__((ext_vector_type(8)));

// Equivariant linear, one irrep segment:
//   out[b, v, i] = c * sum_u w[u, v] * x[b, u, i]
// x/out layout (mul_ir): element (b, u, i) at  b*XDIM + XOFF + u*IRD + i.
//
// Grid: x = ceil(batch/64) M-blocks (rows = consecutive b within one i-plane),
//       y = MUL/64 N-blocks, z = IRD (the i-plane). This keeps every 16-row
//       WMMA tile inside one i-plane: no division in addressing, and all
//       store addresses are base + lane_offset + compile-time immediate.
//
// Each 32-thread block (1 wave) computes a 64x64 output block as a 4x4 grid
// of 16x16 tiles with V_WMMA_F32_16X16X4_F32, K-loop step 4.
template <int MUL, int IRD, int XOFF, int WOFF>
__global__ __launch_bounds__(32) void eqlin_seg_kernel(
    const float* __restrict__ x, const float* __restrict__ w,
    float* __restrict__ out, int batch, float cscale) {
  constexpr int XDIM = 1664;  // 256*1 + 256*3 + 128*5
  const int b0 = blockIdx.x * 64;   // M block (batch rows in this i-plane)
  const int n0 = blockIdx.y * 64;   // N block (output channel v)
  const int i0 = blockIdx.z;        // which component of the irrep
  const int lane = threadIdx.x;     // 0..31
  const int half = lane >> 4;       // 0/1: K pair select (A/B), M upper half (D)
  const int l15 = lane & 15;

  // Per-lane A-fragment row pointers (clamped branchlessly; EXEC stays full).
  const float* aptr[4];
#pragma unroll
  for (int mt = 0; mt < 4; ++mt) {
    int b = b0 + mt * 16 + l15;  // A layout: M = lane&15
    if (b > batch - 1) b = batch - 1;
    aptr[mt] = x + (size_t)b * XDIM + XOFF + i0;  // + k*IRD -> element (b,k)
  }

  // B layout: N = lane&15; regs hold K = ka, ka+1; w[u, v] at WOFF + u*MUL + v
  const float* wcol = w + WOFF + (n0 + l15);

  v8f acc[4][4];
#pragma unroll
  for (int mt = 0; mt < 4; ++mt)
#pragma unroll
    for (int nt = 0; nt < 4; ++nt) acc[mt][nt] = (v8f)0.0f;

  for (int k0 = 0; k0 < MUL; k0 += 4) {
    const int ka = k0 + 2 * half;  // this lane's K pair: ka, ka+1

    v2f a[4];
#pragma unroll
    for (int mt = 0; mt < 4; ++mt) {
      if (IRD == 1) {
        a[mt] = *(const v2f*)(aptr[mt] + ka);  // contiguous along K
      } else {
        a[mt].x = aptr[mt][ka * IRD];
        a[mt].y = aptr[mt][ka * IRD + IRD];
      }
    }

    v2f b[4];
#pragma unroll
    for (int nt = 0; nt < 4; ++nt) {
      const float* p = wcol + ka * MUL + nt * 16;
      b[nt].x = p[0];
      b[nt].y = p[MUL];
    }

#pragma unroll
    for (int mt = 0; mt < 4; ++mt)
#pragma unroll
      for (int nt = 0; nt < 4; ++nt)
        acc[mt][nt] = __builtin_amdgcn_wmma_f32_16x16x4_f32(
            false, a[mt], false, b[nt], (short)0, acc[mt][nt], false, false);
  }

  // ---- Epilogue ----
  // D layout: reg e -> row b = b0 + mt*16 + e + 8*half, col v = n0+nt*16+l15.
  // Uniform base (SGPR) + single lane-varying offset (VGPR) + immediates.
  float* obase = out + (size_t)b0 * XDIM + XOFF + (size_t)n0 * IRD + i0;
  const int lofs = half * 8 * XDIM + l15 * IRD;  // lane-only part

  if (b0 + 64 <= batch) {
    // Interior tile: unguarded stores, all addressing folds to immediates.
#pragma unroll
    for (int mt = 0; mt < 4; ++mt)
#pragma unroll
      for (int nt = 0; nt < 4; ++nt)
#pragma unroll
        for (int e = 0; e < 8; ++e)
          obase[lofs + (mt * 16 + e) * XDIM + nt * 16 * IRD] =
              acc[mt][nt][e] * cscale;
  } else {
    // Boundary tile (rare): guard each row.
#pragma unroll
    for (int mt = 0; mt < 4; ++mt)
#pragma unroll
      for (int nt = 0; nt < 4; ++nt)
#pragma unroll
        for (int e = 0; e < 8; ++e) {
          const int b = b0 + mt * 16 + e + 8 * half;
          if (b < batch)
            obase[lofs + (mt * 16 + e) * XDIM + nt * 16 * IRD] =
                acc[mt][nt][e] * cscale;
        }
  }
}

extern "C" void kernel_launch(void* const* d_in, const int* in_sizes, int n_in,
                              void* d_out, int out_size, void* d_ws,
                              size_t ws_size, hipStream_t stream) {
  const float* x = (const float*)d_in[0];
  const float* w = (const float*)d_in[1];
  float* out = (float*)d_out;

  const int XDIM = 1664;
  const int batch = in_sizes[0] / XDIM;
  const int mblocks = (batch + 63) / 64;

  // Segment 1: 256 x 0e  (MUL=256, IRD=1), xoff=0,    woff=0
  eqlin_seg_kernel<256, 1, 0, 0>
      <<<dim3(mblocks, 4, 1), 32, 0, stream>>>(x, w, out, batch,
                                               0.0625f /* 1/sqrt(256) */);
  // Segment 2: 256 x 1o  (MUL=256, IRD=3), xoff=256,  woff=256*256
  eqlin_seg_kernel<256, 3, 256, 65536>
      <<<dim3(mblocks, 4, 3), 32, 0, stream>>>(x, w, out, batch, 0.0625f);
  // Segment 3: 128 x 2e  (MUL=128, IRD=5), xoff=1024, woff=2*256*256
  eqlin_seg_kernel<128, 5, 1024, 131072>
      <<<dim3(mblocks, 2, 5), 32, 0, stream>>>(
          x, w, out, batch, 0.08838834764831845f /* 1/sqrt(128) */);
}